// BERTRegression_72945724555435
// MI455X (gfx1250) — compile-verified
//
#include <hip/hip_runtime.h>
#include <hip/hip_bf16.h>

typedef unsigned int u32;
typedef float v2f __attribute__((ext_vector_type(2)));
typedef float v4f __attribute__((ext_vector_type(4)));
typedef float v8f __attribute__((ext_vector_type(8)));
typedef u32   v4u __attribute__((ext_vector_type(4)));
typedef int   v4i __attribute__((ext_vector_type(4)));
typedef int   v8i __attribute__((ext_vector_type(8)));

#define Bsz 32
#define Tsz 4096
#define Hsz 256
#define MASK_PENALTY 1000000.0f

#define ROWF 260                  // LDS row stride in floats: 256 + 4 pad (TDM pad)
#define TILE_F (16 * ROWF)        // floats per staged 16x256 tile (padded)
#define TILE_BYTES (TILE_F * 4)   // 16640 B per wave buffer

// ---------------------------------------------------------------------------
// Kernel 1: build c[t] = we*w_ema[t] + wm/T, reduce sumW, zero accumulators.
// ---------------------------------------------------------------------------
__global__ __launch_bounds__(256) void prep_kernel(
    const float* __restrict__ we_p, const float* __restrict__ wm_p,
    const float* __restrict__ Wv,
    float* __restrict__ c, float* __restrict__ acc,
    float* __restrict__ pensum, float* __restrict__ misc) {
  const int t = threadIdx.x;
  const float alpha = 0.1f, decay = 0.9f;
  const float wE = we_p[0], wM = wm_p[0];
  #pragma unroll
  for (int k = 0; k < Tsz / 256; ++k) {
    int tt = t + 256 * k;
    float p = __powf(decay, (float)(Tsz - 1 - tt));   // underflows to 0 like f32 ref
    float w = (tt == 0) ? p : alpha * p;
    c[tt] = wE * w + wM * (1.0f / (float)Tsz);
  }
  __shared__ float red[256];
  red[t] = Wv[t];                                     // H == 256 == blockDim
  __syncthreads();
  for (int s = 128; s >= 1; s >>= 1) {
    if (t < s) red[t] += red[t + s];
    __syncthreads();
  }
  if (t == 0) misc[0] = red[0];                       // sum of W
  if (t < Bsz) { acc[t] = 0.0f; pensum[t] = 0.0f; }   // fresh every call
}

// ---------------------------------------------------------------------------
// Kernel 2: pensum[b] = sum_t c[t] * (1 - mask[b,t])
// ---------------------------------------------------------------------------
__global__ __launch_bounds__(256) void pensum_kernel(
    const int* __restrict__ mask, const float* __restrict__ c,
    float* __restrict__ pensum) {
  const int b = blockIdx.x, t = threadIdx.x;
  float s = 0.0f;
  for (int i = t; i < Tsz; i += 256)
    s += c[i] * (float)(1 - mask[b * Tsz + i]);
  __shared__ float red[256];
  red[t] = s;
  __syncthreads();
  for (int sft = 128; sft >= 1; sft >>= 1) {
    if (t < sft) red[t] += red[t + sft];
    __syncthreads();
  }
  if (t == 0) pensum[b] = red[0];
}

// ---------------------------------------------------------------------------
// Kernel 3: main streaming GEMV.
//   TDM (tensor_load_to_lds) DMAs each wave's 16x256 f32 tile into LDS with
//   4-DWORD row padding (row stride 260 floats -> conflict-free ds_load_b128),
//   then a V_WMMA_F32_16X16X4_F32 chain computes the 16 row dot-products.
//   K-column mapping per b128: lane<16 -> cols {8i..8i+3}, lane>=16 ->
//   cols {8i+4..8i+7}; A and B use the same mapping so the dot is exact.
// ---------------------------------------------------------------------------
__global__ __launch_bounds__(256) void gemv_tdm_wmma_kernel(
    const float* __restrict__ x, const float* __restrict__ Wv,
    const float* __restrict__ c, float* __restrict__ acc) {
  extern __shared__ float lds[];            // 8 * TILE_F floats (dynamic)
  __shared__ float sW[Hsz];
  const int tid = threadIdx.x;
  sW[tid] = Wv[tid];
  __syncthreads();

  const int lane = tid & 31;
  const int wave_v = tid >> 5;
  const u32 wave = (u32)__builtin_amdgcn_readfirstlane(wave_v);   // SGPR
  const int tile = __builtin_amdgcn_readfirstlane((int)(blockIdx.x * 8u + wave));
  const long r0 = (long)tile * 16;          // global row = b*T + t0
  const int b  = (int)(r0 >> 12);
  const int t0 = (int)(r0 & (Tsz - 1));

  float* buf = lds + wave * TILE_F;
  const u32 lds_byte = (u32)(size_t)buf;    // byte offset within LDS
  const unsigned long long ga = (unsigned long long)(const void*)(x + r0 * Hsz);

  // ---- Tensor DMA descriptor (D#), 2D tile 16 rows x 256 f32 ----
  v4u g0;
  g0[0] = 1u;                                           // count=1, user mode
  g0[1] = lds_byte;                                     // lds_addr
  g0[2] = (u32)(ga & 0xFFFFFFFFull);                    // global_addr[31:0]
  g0[3] = (u32)((ga >> 32) & 0x1FFFFFFull) | (2u << 30); // addr[56:32] | type=2

  v8i g1;
  g1[0] = (int)((2u << 16)      // data_size = 4 bytes
              | (1u << 20)      // pad_enable
              | (7u << 22)      // pad_interval: every 256 DWORDs (one row)
              | (3u << 25));    // pad_amount: 4 DWORDs -> row stride 260 f
  g1[1] = (int)(256u << 16);    // tensor_dim0 = 256 (low 16)
  g1[2] = (int)(16u << 16);     // tensor_dim0 hi=0 | tensor_dim1 = 16
  g1[3] = (int)(256u << 16);    // tensor_dim1 hi=0 | tile_dim0 = 256
  g1[4] = 16;                   // tile_dim1 = 16 | tile_dim2 = 0
  g1[5] = 256;                  // tensor_dim0_stride = 256 elements
  g1[6] = 0;
  g1[7] = 0;
  v4i g2 = {0, 0, 0, 0};        // dims 2..4 unused
  v4i g3 = {0, 0, 0, 0};
  v8i g4 = {0, 0, 0, 0, 0, 0, 0, 0};  // extra operand on 6-arg toolchain

  __builtin_amdgcn_tensor_load_to_lds(g0, g1, g2, g3, g4, 0);
  __builtin_amdgcn_s_wait_tensorcnt((short)0);          // tile resident in LDS

  // ---- consume: 32 x (ds_load_b128 + 2 WMMA) ----
  const int half = lane >> 4;               // 0: cols 8i..8i+3, 1: 8i+4..8i+7
  const int m    = lane & 15;
  const float* ap = buf + m * ROWF + half * 4;
  const float* wp = sW + half * 4;

  v8f C = {0.f, 0.f, 0.f, 0.f, 0.f, 0.f, 0.f, 0.f};
  #pragma unroll 8
  for (int i = 0; i < Hsz / 8; ++i) {
    v4f av = *(const v4f*)(ap + 8 * i);     // ds_load_b128, conflict-free
    v4f wv = *(const v4f*)(wp + 8 * i);     // broadcast within half-wave
    v2f a0 = {av[0], av[1]}, a1 = {av[2], av[3]};
    v2f b0 = {wv[0], wv[1]}, b1 = {wv[2], wv[3]};
    C = __builtin_amdgcn_wmma_f32_16x16x4_f32(false, a0, false, b0,
                                              (short)0, C, false, false);
    C = __builtin_amdgcn_wmma_f32_16x16x4_f32(false, a1, false, b1,
                                              (short)0, C, false, false);
  }

  // Lane holds s[row] for rows (half ? 8..15 : 0..7) in C[0..7], replicated
  // 16x over N -> scale by 1/16 before the full-wave reduction.
  const int bt = t0 + half * 8;
  float partial = 0.0f;
  #pragma unroll
  for (int r = 0; r < 8; ++r) partial += C[r] * c[bt + r];
  partial *= (1.0f / 16.0f);
  #pragma unroll
  for (int off = 16; off >= 1; off >>= 1) partial += __shfl_xor(partial, off, 32);
  if (lane == 0) atomicAdd(&acc[b], partial);
}

// ---------------------------------------------------------------------------
// Kernel 4: out[b] = acc[b] - 1e6 * sumW * pensum[b] + bias
// ---------------------------------------------------------------------------
__global__ void finalize_kernel(const float* __restrict__ acc,
                                const float* __restrict__ pensum,
                                const float* __restrict__ misc,
                                const float* __restrict__ bias,
                                float* __restrict__ out) {
  int b = threadIdx.x;
  if (b < Bsz)
    out[b] = acc[b] - MASK_PENALTY * misc[0] * pensum[b] + bias[0];
}

extern "C" void kernel_launch(void* const* d_in, const int* in_sizes, int n_in,
                              void* d_out, int out_size, void* d_ws, size_t ws_size,
                              hipStream_t stream) {
  (void)in_sizes; (void)n_in; (void)out_size; (void)ws_size;
  const float* x    = (const float*)d_in[0];  // [B,T,H] f32
  const int*   mask = (const int*)d_in[1];    // [B,T] i32
  const float* we   = (const float*)d_in[2];  // [1]
  const float* wm   = (const float*)d_in[3];  // [1]
  const float* Wv   = (const float*)d_in[4];  // [1,H]
  const float* bias = (const float*)d_in[5];  // [1]
  float* out = (float*)d_out;                 // [B]

  float* ws     = (float*)d_ws;
  float* c      = ws;                         // [4096]
  float* acc    = ws + Tsz;                   // [32]
  float* pensum = ws + Tsz + Bsz;             // [32]
  float* misc   = ws + Tsz + 2 * Bsz;         // [1] sumW

  prep_kernel<<<1, 256, 0, stream>>>(we, wm, Wv, c, acc, pensum, misc);
  pensum_kernel<<<Bsz, 256, 0, stream>>>(mask, c, pensum);
  gemv_tdm_wmma_kernel<<<(Bsz * Tsz / 16) / 8, 256, 8 * TILE_BYTES, stream>>>(
      x, Wv, c, acc);
  finalize_kernel<<<1, 32, 0, stream>>>(acc, pensum, misc, bias, out);
}